// LightGCN_28415503630676
// MI455X (gfx1250) — compile-verified
//
#include <hip/hip_runtime.h>

#define N_USERS_C  100000
#define N_ITEMS_C  50000
#define N_NODES_C  150000
#define DIM        64
#define N_LAYERS_C 3

// ---- CDNA5 async global->LDS support (probe via __has_builtin) -------------
#if defined(__has_builtin)
#if __has_builtin(__builtin_amdgcn_global_load_async_to_lds_b128)
#define HAVE_ASYNC_LDS 1
#endif
#if __has_builtin(__builtin_amdgcn_s_wait_asynccnt)
#define HAVE_WAIT_ASYNC 1
#endif
#endif

typedef int v4i __attribute__((ext_vector_type(4)));
typedef __attribute__((address_space(1))) v4i* as1_v4i_ptr;
typedef __attribute__((address_space(3))) v4i* as3_v4i_ptr;

__device__ __forceinline__ void async_load16(const void* gsrc, void* ldst) {
#ifdef HAVE_ASYNC_LDS
  __builtin_amdgcn_global_load_async_to_lds_b128(
      (as1_v4i_ptr)(void*)(size_t)(const char*)gsrc,
      (as3_v4i_ptr)ldst,
      /*offset=*/0, /*cpol=*/0);
#else
  *(int4*)ldst = *(const int4*)gsrc;   // synchronous fallback
#endif
}

__device__ __forceinline__ void wait_async_all() {
#ifdef HAVE_ASYNC_LDS
#ifdef HAVE_WAIT_ASYNC
  __builtin_amdgcn_s_wait_asynccnt(0);
#else
  asm volatile("s_wait_asynccnt 0x0" ::: "memory");
#endif
#endif
}

__device__ __forceinline__ void wait_async_le3() {
#ifdef HAVE_ASYNC_LDS
#ifdef HAVE_WAIT_ASYNC
  __builtin_amdgcn_s_wait_asynccnt(3);
#else
  asm volatile("s_wait_asynccnt 0x3" ::: "memory");
#endif
#endif
}

// Native no-return f32 atomic add (global_atomic_add_f32, STOREcnt-tracked).
__device__ __forceinline__ void atomic_add_f32(float* p, float v) {
  unsafeAtomicAdd(p, v);
}

// ---- SpMM scatter: nxt[row] += vals * cur[col] -----------------------------
// FIRST function in the module so the disasm snippet shows the hot kernel.
// Each wave double-buffers its own 128-edge chunk (row/col/vals) in LDS via
// async b128 loads, then 32 lanes * float2 cover the 64-dim row per edge.
#define WAVES_PER_BLOCK 8
#define CHUNK 128

__global__ __launch_bounds__(256) void lgcn_scatter_kernel(
    const int* __restrict__ row, const int* __restrict__ col,
    const float* __restrict__ vals, const float* __restrict__ cur,
    float* __restrict__ nxt, int n_edges)
{
  __shared__ int   s_row[2][WAVES_PER_BLOCK][CHUNK];
  __shared__ int   s_col[2][WAVES_PER_BLOCK][CHUNK];
  __shared__ float s_val[2][WAVES_PER_BLOCK][CHUNK];

  const int lane = threadIdx.x & 31;
  const int w    = threadIdx.x >> 5;
  const int gw   = blockIdx.x * WAVES_PER_BLOCK + w;   // global wave id
  const int nw   = gridDim.x * WAVES_PER_BLOCK;        // total waves
  const int full_chunks = n_edges >> 7;

  auto issue = [&](int buf, int c) {
    // 128 edges * 4B = 512B per array = 32 lanes * 16B, all 16B-aligned
    async_load16(row  + (size_t)c * CHUNK + lane * 4, &s_row[buf][w][lane * 4]);
    async_load16(col  + (size_t)c * CHUNK + lane * 4, &s_col[buf][w][lane * 4]);
    async_load16(vals + (size_t)c * CHUNK + lane * 4, &s_val[buf][w][lane * 4]);
  };

  const float* curl = cur + lane * 2;
  float*       nxtl = nxt + lane * 2;

  int buf = 0;
  int c = gw;
  if (c < full_chunks) issue(0, c);
  for (; c < full_chunks; c += nw) {
    int cn = c + nw;
    if (cn < full_chunks) { issue(buf ^ 1, cn); wait_async_le3(); }
    else                  { wait_async_all(); }

    const int*   rp = s_row[buf][w];
    const int*   cp = s_col[buf][w];
    const float* vp = s_val[buf][w];
#pragma unroll 4
    for (int e = 0; e < CHUNK; ++e) {
      int   r  = rp[e];
      int   cc = cp[e];
      float v  = vp[e];
      const float2 x = *(const float2*)(curl + (size_t)cc * DIM);
      float* dst = nxtl + (size_t)r * DIM;
      atomic_add_f32(dst,     v * x.x);
      atomic_add_f32(dst + 1, v * x.y);
    }
    buf ^= 1;
  }

  // tail (n_edges not multiple of 128) — handled by one wave, direct loads
  if (blockIdx.x == 0 && w == 0) {
    for (int e = full_chunks << 7; e < n_edges; ++e) {
      int   r  = row[e];
      int   cc = col[e];
      float v  = vals[e];
      const float2 x = *(const float2*)(curl + (size_t)cc * DIM);
      float* dst = nxtl + (size_t)r * DIM;
      atomic_add_f32(dst,     v * x.x);
      atomic_add_f32(dst + 1, v * x.y);
    }
  }
}

// ---- init: cur = concat(user_emb, item_emb); nxt = 0 -----------------------
__global__ __launch_bounds__(256) void lgcn_init_kernel(
    const float4* __restrict__ uemb, const float4* __restrict__ iemb,
    float4* __restrict__ cur, float4* __restrict__ nxt,
    int n4_users, int n4_total)
{
  int i = blockIdx.x * blockDim.x + threadIdx.x;
  if (i >= n4_total) return;
  float4 v = (i < n4_users) ? uemb[i] : iemb[i - n4_users];
  cur[i] = v;
  nxt[i] = make_float4(0.f, 0.f, 0.f, 0.f);
}

// ---- zero a node buffer ----------------------------------------------------
__global__ __launch_bounds__(256) void lgcn_zero_kernel(float4* __restrict__ p, int n4)
{
  int i = blockIdx.x * blockDim.x + threadIdx.x;
  if (i < n4) p[i] = make_float4(0.f, 0.f, 0.f, 0.f);
}

// ---- gather batch rows, scale by 0.25, write or accumulate into d_out ------
__global__ __launch_bounds__(256) void lgcn_gather_kernel(
    const float* __restrict__ src,
    const int* __restrict__ users, const int* __restrict__ pos,
    const int* __restrict__ neg, float* __restrict__ out,
    int batch, int write_mode)
{
  int tid  = blockIdx.x * blockDim.x + threadIdx.x;
  int lane = tid & 31;
  int b    = (tid >> 5) % batch;
  int s    = tid / (batch * 32);
  if (s >= 3) return;
  int node = (s == 0) ? users[b]
                      : (((s == 1) ? pos[b] : neg[b]) + N_USERS_C);
  const float2 x = *(const float2*)(src + (size_t)node * DIM + lane * 2);
  float2* o = (float2*)out + (size_t)tid;   // ((s*batch+b)*32 + lane)-th float2
  float2 v = make_float2(0.25f * x.x, 0.25f * x.y);
  if (write_mode) {
    *o = v;
  } else {
    float2 old = *o;
    *o = make_float2(old.x + v.x, old.y + v.y);
  }
}

extern "C" void kernel_launch(void* const* d_in, const int* in_sizes, int n_in,
                              void* d_out, int out_size, void* d_ws, size_t ws_size,
                              hipStream_t stream)
{
  const float* uemb  = (const float*)d_in[0];
  const float* iemb  = (const float*)d_in[1];
  const int*   row   = (const int*)d_in[2];
  const int*   col   = (const int*)d_in[3];
  const float* vals  = (const float*)d_in[4];
  const int*   users = (const int*)d_in[5];
  const int*   pos   = (const int*)d_in[6];
  const int*   neg   = (const int*)d_in[7];
  const int n_edges  = in_sizes[2];
  const int batch    = in_sizes[5];

  const size_t S = (size_t)N_NODES_C * DIM;   // floats per node buffer
  float* A = (float*)d_ws;                    // ping
  float* B = A + S;                           // pong   (2 * 38.4 MB used)

  const int n4       = (int)(S / 4);
  const int n4_users = N_USERS_C * DIM / 4;

  lgcn_init_kernel<<<(n4 + 255) / 256, 256, 0, stream>>>(
      (const float4*)uemb, (const float4*)iemb, (float4*)A, (float4*)B,
      n4_users, n4);

  const int gth = 3 * batch * 32;
  lgcn_gather_kernel<<<(gth + 255) / 256, 256, 0, stream>>>(
      A, users, pos, neg, (float*)d_out, batch, /*write=*/1);

  float* cur = A;
  float* nxt = B;
  for (int l = 0; l < N_LAYERS_C; ++l) {
    lgcn_scatter_kernel<<<1024, 256, 0, stream>>>(row, col, vals, cur, nxt, n_edges);
    lgcn_gather_kernel<<<(gth + 255) / 256, 256, 0, stream>>>(
        nxt, users, pos, neg, (float*)d_out, batch, /*write=*/0);
    if (l + 1 < N_LAYERS_C) {
      lgcn_zero_kernel<<<(n4 + 255) / 256, 256, 0, stream>>>((float4*)cur, n4);
      float* t = cur; cur = nxt; nxt = t;
    }
  }
}